// GraphAttentionLayer_75196287418915
// MI455X (gfx1250) — compile-verified
//
#include <hip/hip_runtime.h>
#include <hip/hip_bf16.h>
#include <math.h>

#define N_NODES 16384
#define N_EDGES 131072
#define DIM     64
#define DIME    32
#define HEADS   8
#define HD      512           // HEADS*DIM
#define INV_NORM (1.0f/160.0f) // 1/(2*D+DE)
#define ALPHA_SLOPE 0.2f

typedef __attribute__((ext_vector_type(2))) float v2f;
typedef __attribute__((ext_vector_type(8))) float v8f;

__device__ __forceinline__ v8f wmma_f32(v2f a, v2f b, v8f c) {
    // V_WMMA_F32_16X16X4_F32 : D = A(16x4) * B(4x16) + C(16x16), all fp32
    return __builtin_amdgcn_wmma_f32_16x16x4_f32(
        /*neg_a=*/false, a, /*neg_b=*/false, b,
        /*c_mod=*/(short)0, c, /*reuse_a=*/false, /*reuse_b=*/false);
}

// ---------------------------------------------------------------------------
// Generic fp32 WMMA GEMM: C[M x Nc] = A[M x K] @ B[K x Nc] + bias[Nc]
// One wave computes a 16-row x 64-col strip (4 WMMA tiles, shared A fragment).
// A-fragment (16x4): lane<16 holds A[m][k..k+1], lane>=16 holds A[m][k+2..k+3]
// B-fragment (4x16): lane<16 holds B[k][n],B[k+1][n]; lane>=16 B[k+2][n],B[k+3][n]
// D (16x16): VGPR r, lane l -> (M = r + 8*(l>>4), N = l&15)
// ---------------------------------------------------------------------------
__global__ void gemm_wmma_kernel(const float* __restrict__ A,
                                 const float* __restrict__ B,
                                 const float* __restrict__ bias,
                                 float* __restrict__ C,
                                 int M, int K, int Ncols) {
    const int lane  = threadIdx.x & 31;
    const int wave  = threadIdx.x >> 5;
    const int gw    = blockIdx.x * (blockDim.x >> 5) + wave;
    const int strips = Ncols >> 6;             // 64-wide column strips
    const int row_tile = gw / strips;
    const int strip    = gw % strips;
    const int m0 = row_tile * 16;
    if (m0 >= M) return;
    const int n0 = strip * 64;

    const int mlane = lane & 15;
    const int khalf = lane >> 4;               // 0 or 1 -> K pair select

    v8f acc0 = {}, acc1 = {}, acc2 = {}, acc3 = {};

    const float* Arow = A + (size_t)(m0 + mlane) * K;
    for (int kk = 0; kk < K; kk += 4) {
        const int kb = kk + khalf * 2;
        const v2f a = *(const v2f*)(Arow + kb);       // 8B aligned (kb even)

        const float* Brow0 = B + (size_t)kb * Ncols + n0 + mlane;
        const float* Brow1 = Brow0 + Ncols;
        v2f b0, b1, b2, b3;
        b0.x = Brow0[0];   b0.y = Brow1[0];
        b1.x = Brow0[16];  b1.y = Brow1[16];
        b2.x = Brow0[32];  b2.y = Brow1[32];
        b3.x = Brow0[48];  b3.y = Brow1[48];

        acc0 = wmma_f32(a, b0, acc0);
        acc1 = wmma_f32(a, b1, acc1);
        acc2 = wmma_f32(a, b2, acc2);
        acc3 = wmma_f32(a, b3, acc3);
    }

    const int mbase = m0 + khalf * 8;
    const int ncol  = mlane;
    const float bb0 = bias[n0 + ncol];
    const float bb1 = bias[n0 + 16 + ncol];
    const float bb2 = bias[n0 + 32 + ncol];
    const float bb3 = bias[n0 + 48 + ncol];
#pragma unroll
    for (int r = 0; r < 8; ++r) {
        float* Crow = C + (size_t)(mbase + r) * Ncols + n0 + ncol;
        Crow[0]  = acc0[r] + bb0;
        Crow[16] = acc1[r] + bb1;
        Crow[32] = acc2[r] + bb2;
        Crow[48] = acc3[r] + bb3;
    }
}

// hs[n][head] = sum_d h[n][head*64 + d]
__global__ void rowsum_kernel(const float* __restrict__ h, float* __restrict__ hs) {
    const int idx = blockIdx.x * blockDim.x + threadIdx.x;   // n*8 + head
    if (idx >= N_NODES * HEADS) return;
    const int n = idx >> 3, hh = idx & 7;
    const float* p = h + (size_t)n * HD + hh * DIM;
    float s = 0.f;
#pragma unroll 8
    for (int i = 0; i < DIM; ++i) s += p[i];
    hs[idx] = s;
}

// ksum[k*8+hh] = sum_j kernel_edge[k][hh*32+j];  bsum[hh] = sum_j bias_edge[hh*32+j]
__global__ void ksum_kernel(const float* __restrict__ ke,
                            const float* __restrict__ be,
                            float* __restrict__ ksum, float* __restrict__ bsum) {
    const int t = threadIdx.x;                 // 256 threads: (k,hh)
    const int k = t >> 3, hh = t & 7;
    const float* row = ke + (size_t)k * (DIME * HEADS) + hh * DIME;
    float s = 0.f;
#pragma unroll
    for (int j = 0; j < DIME; ++j) s += row[j];
    ksum[t] = s;
    if (t < HEADS) {
        float b = 0.f;
        const float* bp = be + t * DIME;
#pragma unroll
        for (int j = 0; j < DIME; ++j) b += bp[j];
        bsum[t] = b;
    }
}

// esum[e*8+hh] = edge_features[e] . ksum[:,hh] + bsum[hh]
__global__ void esum_kernel(const float* __restrict__ ef,
                            const float* __restrict__ ksum,
                            const float* __restrict__ bsum,
                            float* __restrict__ esum) {
    __shared__ float ks[DIME * HEADS];
    __shared__ float bs[HEADS];
    if (threadIdx.x < DIME * HEADS) ks[threadIdx.x] = ksum[threadIdx.x];
    if (threadIdx.x < HEADS)        bs[threadIdx.x] = bsum[threadIdx.x];
    __syncthreads();
    const int idx = blockIdx.x * blockDim.x + threadIdx.x;   // e*8 + hh
    if (idx >= N_EDGES * HEADS) return;
    const int e = idx >> 3, hh = idx & 7;
    const float* fe = ef + (size_t)e * DIME;
    float s = bs[hh];
#pragma unroll
    for (int k = 0; k < DIME; ++k) s += fe[k] * ks[k * 8 + hh];
    esum[idx] = s;
}

// recv is sorted and every node appears -> boundary detect builds CSR offsets.
__global__ void rowoff_kernel(const int* __restrict__ pair, int* __restrict__ row_start) {
    const int e = blockIdx.x * blockDim.x + threadIdx.x;
    if (e >= N_EDGES) return;
    const int r = pair[2 * e];
    if (e == 0 || pair[2 * (e - 1)] != r) row_start[r] = e;
    if (e == 0) row_start[N_NODES] = N_EDGES;
}

// One 256-thread block per node. LDS-stage exp-weights per 32-edge chunk, then
// each thread owns (head = tid>>5, 2 feature slots) and accumulates numer/denom.
__global__ void aggregate_kernel(const float* __restrict__ h,
                                 const float* __restrict__ hs,
                                 const float* __restrict__ esum,
                                 const int* __restrict__ pair,
                                 const int* __restrict__ row_start,
                                 float* __restrict__ agg) {
    __shared__ float exbuf[32 * HEADS];
    __shared__ int   sendbuf[32];
    const int n    = blockIdx.x;
    const int s    = row_start[n];
    const int eend = row_start[n + 1];
    const int tid  = threadIdx.x;
    const int head = tid >> 5;
    const int i2   = tid & 31;

    float accx = 0.f, accy = 0.f, dn = 0.f;

    for (int base = s; base < eend; base += 32) {
        const int cnt = min(32, eend - base);
        __syncthreads();                       // protect exbuf reuse
        if (tid < cnt * HEADS) {
            const int j = tid >> 3, hh = tid & 7;
            const int eidx = base + j;
            const int snd  = pair[2 * eidx + 1];
            const float a  = (hs[n * HEADS + hh] + hs[snd * HEADS + hh] +
                              esum[eidx * HEADS + hh]) * INV_NORM;
            const float lr = a > 0.f ? a : ALPHA_SLOPE * a;
            exbuf[j * HEADS + hh] = __expf(lr);
            if (hh == 0) sendbuf[j] = snd;
        }
        __syncthreads();
        for (int j = 0; j < cnt; ++j) {
            const float exv = exbuf[j * HEADS + head];
            dn += exv;
            const float* hp = h + (size_t)sendbuf[j] * HD + head * DIM + i2 * 2;
            accx += exv * hp[0];
            accy += exv * hp[1];
        }
    }

    const float inv = 1.f / dn;
    const size_t o = (size_t)n * HD + head * DIM + i2 * 2;
    agg[o]     = accx * inv;
    agg[o + 1] = accy * inv;
}

extern "C" void kernel_launch(void* const* d_in, const int* in_sizes, int n_in,
                              void* d_out, int out_size, void* d_ws, size_t ws_size,
                              hipStream_t stream) {
    const float* atom_features = (const float*)d_in[0];   // (N,64)
    const float* edge_features = (const float*)d_in[1];   // (E,32)
    const int*   pair_indices  = (const int*)d_in[2];     // (E,2) [recv,send]
    const float* kernel_       = (const float*)d_in[3];   // (64,512)
    const float* bias_         = (const float*)d_in[4];   // (512)
    const float* kernel_edge   = (const float*)d_in[5];   // (32,256)
    const float* bias_edge     = (const float*)d_in[6];   // (256)
    const float* kernel_wo     = (const float*)d_in[7];   // (512,64)
    const float* bias_wo       = (const float*)d_in[8];   // (64)
    float* out = (float*)d_out;                           // (N,64)

    // workspace layout
    char* ws = (char*)d_ws;
    float* h    = (float*)(ws);                                   // 32 MB
    float* agg  = (float*)(ws + (size_t)33554432);                // 32 MB
    float* hs   = (float*)(ws + (size_t)67108864);                // 512 KB
    float* esum = (float*)(ws + (size_t)67633152);                // 4 MB
    float* ksum = (float*)(ws + (size_t)71827456);                // 1 KB
    float* bsum = (float*)(ws + (size_t)71828480);                // 32 B
    int*   roff = (int*)  (ws + (size_t)71828512);                // (N+1)*4 B

    // 1) h = x @ kernel + bias   (WMMA fp32, 16x64 strip per wave)
    {
        const int waves = (N_NODES / 16) * (HD / 64);   // 8192
        gemm_wmma_kernel<<<waves / 8, 256, 0, stream>>>(
            atom_features, kernel_, bias_, h, N_NODES, DIM, HD);
    }
    // 2) per-(node,head) rowsums of h
    rowsum_kernel<<<(N_NODES * HEADS) / 256, 256, 0, stream>>>(h, hs);
    // 3) fold kernel_edge columns per head
    ksum_kernel<<<1, 256, 0, stream>>>(kernel_edge, bias_edge, ksum, bsum);
    // 4) per-(edge,head) logit contribution from edge features
    esum_kernel<<<(N_EDGES * HEADS) / 256, 256, 0, stream>>>(
        edge_features, ksum, bsum, esum);
    // 5) CSR offsets from sorted recv
    rowoff_kernel<<<N_EDGES / 256, 256, 0, stream>>>(pair_indices, roff);
    // 6) softmax-weighted aggregation (no atomics; one block per node)
    aggregate_kernel<<<N_NODES, 256, 0, stream>>>(
        h, hs, esum, pair_indices, roff, agg);
    // 7) out = agg @ kernel_wo + bias_wo  (WMMA fp32)
    {
        const int waves = (N_NODES / 16) * (DIM / 64);  // 1024
        gemm_wmma_kernel<<<waves / 8, 256, 0, stream>>>(
            agg, kernel_wo, bias_wo, out, N_NODES, HD, DIM);
    }
}